// QuantumCircuit_31241592111324
// MI455X (gfx1250) — compile-verified
//
#include <hip/hip_runtime.h>

// 24-qubit state-vector simulator for MI455X (gfx1250).
// State = 2^24 f32 = 64MB (L2-resident). Per block:
//   - RY layer = Kron product of 24 RY(2x2) = six 16x16 group matrices,
//     applied with V_WMMA_F32_16X16X4_F32 (4 chained K=4 accumulations).
//   - CNOT chain = basis permutation x = y ^ (y>>1), folded into the next
//     pass's read addressing (and the final squaring pass).

typedef __attribute__((ext_vector_type(2))) float v2f;
typedef __attribute__((ext_vector_type(8))) float v8f;

#define NQ 24
#define NSTATE (1u << NQ)
#define ROWLEN 4096   // 2^12

// Build three 16x16 group matrices (groups g_base..g_base+2) into LDS.
// mats[st*256 + i*16 + j]; stage st uses contraction stride {1,16,256} and
// group g = g_base + st (group g covers flat bits [4g+3:4g] = qubits 23-4g-t).
__device__ __forceinline__ void build_mats(const float* __restrict__ angles,
                                           int blk, int g_base,
                                           float* __restrict__ mats) {
  for (int e = threadIdx.x; e < 3 * 256; e += blockDim.x) {
    int st = e >> 8;
    int ij = e & 255;
    int i = ij >> 4, j = ij & 15;
    int g = g_base + st;
    float p = 1.0f;
#pragma unroll
    for (int t = 0; t < 4; ++t) {
      int q = 23 - (4 * g + t);          // qubit index for bit t of this group
      float half = 0.5f * angles[blk * NQ + q];
      float c = __builtin_cosf(half);
      float s = __builtin_sinf(half);
      int ib = (i >> t) & 1, jb = (j >> t) & 1;
      // RY = [[c,-s],[s,c]] : m[ib][jb]
      float f = (ib == jb) ? c : (ib ? s : -s);
      p *= f;
    }
    mats[e] = p;
  }
  __syncthreads();
}

// Apply 16x16 matrix M (row-major in LDS) along the stride-S axis of a
// 4096-float LDS buffer, in place, using V_WMMA_F32_16X16X4_F32.
// 256 columns processed as 16 matmuls of (16x16)x(16x16).
template <int S>
__device__ __forceinline__ void apply16_stage(const float* __restrict__ M,
                                              float* __restrict__ buf) {
  const int lane = threadIdx.x & 31;   // wave32
  const int row  = lane & 15;
  const int hi   = lane >> 4;

  // A operand: lane holds matrix row `row`; VGPR v of chunk kk holds K = 4kk + v + 2*hi.
  v2f A[4];
#pragma unroll
  for (int kk = 0; kk < 4; ++kk) {
    int k0 = 4 * kk + 2 * hi;
    A[kk].x = M[row * 16 + k0];
    A[kk].y = M[row * 16 + k0 + 1];
  }

  for (int g = 0; g < 16; ++g) {
    const int c = g * 16 + row;        // this lane's column
    int base;
    if (S == 1)       base = c << 4;
    else if (S == 16) base = ((c & 0xF0) << 4) | (c & 0x0F);
    else              base = c;        // S == 256

    v8f acc = {};
#pragma unroll
    for (int kk = 0; kk < 4; ++kk) {
      int k0 = 4 * kk + 2 * hi;
      v2f B;
      B.x = buf[base + (k0    ) * S];
      B.y = buf[base + (k0 + 1) * S];
      acc = __builtin_amdgcn_wmma_f32_16x16x4_f32(
          false, A[kk], false, B, (short)0, acc, false, false);
    }
    // D layout: VGPR r holds result row (r + 8*hi), column c.
#pragma unroll
    for (int r = 0; r < 8; ++r) {
      buf[base + (r + 8 * hi) * S] = acc[r];
    }
  }
}

__device__ __forceinline__ void apply_three_stages(const float* mats, float* b) {
  apply16_stage<1>(mats, b);
  apply16_stage<16>(mats + 256, b);
  apply16_stage<256>(mats + 512, b);
}

// Pass over the low 12 index bits (groups 0..2). One 16KB row per wave.
// If apply_perm, fold the previous block's CNOT permutation into the gather:
// new[H*4096+l] reads old[(H^(H>>1))*4096 + (l ^ (l>>1) ^ ((H&1)<<11))].
__global__ void __launch_bounds__(64) pass_low(const float* __restrict__ in,
                                               float* __restrict__ out,
                                               const float* __restrict__ angles,
                                               int blk, int apply_perm) {
  __shared__ float mats[768];
  __shared__ float buf[2][ROWLEN];
  build_mats(angles, blk, 0, mats);

  const int wave = threadIdx.x >> 5;
  const int lane = threadIdx.x & 31;
  const int H    = blockIdx.x * 2 + wave;
  const int Hs   = apply_perm ? (H ^ (H >> 1)) : H;
  const int hbit = apply_perm ? ((H & 1) << 11) : 0;
  const float* src = in + (size_t)Hs * ROWLEN;
  float* b = buf[wave];

  if (apply_perm) {
    for (int t = lane; t < ROWLEN; t += 32)
      b[t] = src[t ^ (t >> 1) ^ hbit];   // stays inside one 128B line window
  } else {
    for (int t = lane; t < ROWLEN; t += 32)
      b[t] = src[t];
  }
  // same-wave DS ordering: stores complete before compute-phase loads
  apply_three_stages(mats, b);

  float* dst = out + (size_t)H * ROWLEN;
  for (int t = lane; t < ROWLEN; t += 32) dst[t] = b[t];
}

// Pass over the high 12 index bits (groups 3..5). Workgroup handles 2 adjacent
// columns: cooperative float2 loads (8B granularity), one column per wave.
__global__ void __launch_bounds__(64) pass_high(const float* __restrict__ in,
                                                float* __restrict__ out,
                                                const float* __restrict__ angles,
                                                int blk) {
  __shared__ float mats[768];
  __shared__ float buf[2][ROWLEN];
  build_mats(angles, blk, 3, mats);

  const int L0 = blockIdx.x * 2;
  for (int h = threadIdx.x; h < ROWLEN; h += 64) {
    const float2 v = *(const float2*)(in + (size_t)h * ROWLEN + L0);
    buf[0][h] = v.x;
    buf[1][h] = v.y;
  }
  __syncthreads();

  apply_three_stages(mats, buf[threadIdx.x >> 5]);
  __syncthreads();

  for (int h = threadIdx.x; h < ROWLEN; h += 64) {
    float2 v;
    v.x = buf[0][h];
    v.y = buf[1][h];
    *(float2*)(out + (size_t)h * ROWLEN + L0) = v;
  }
}

__global__ void init_kernel(float* __restrict__ st) {
  unsigned i = blockIdx.x * blockDim.x + threadIdx.x;
  st[i] = (i == 0) ? 1.0f : 0.0f;
}

// Final CNOT-chain permutation + |psi|^2: out[y] = state[y ^ (y>>1)]^2.
__global__ void finalize_kernel(const float* __restrict__ st,
                                float* __restrict__ out) {
  unsigned y = blockIdx.x * blockDim.x + threadIdx.x;
  unsigned x = y ^ (y >> 1);
  float v = st[x];
  out[y] = v * v;
}

extern "C" void kernel_launch(void* const* d_in, const int* in_sizes, int n_in,
                              void* d_out, int out_size, void* d_ws, size_t ws_size,
                              hipStream_t stream) {
  (void)in_sizes; (void)n_in; (void)out_size; (void)ws_size;
  const float* angles = (const float*)d_in[0];   // [4][24] f32
  float* X = (float*)d_ws;    // 64MB state buffer
  float* Y = (float*)d_out;   // reuse output buffer as inter-pass scratch

  init_kernel<<<NSTATE / 256, 256, 0, stream>>>(X);
  for (int b = 0; b < 4; ++b) {
    pass_low <<<ROWLEN / 2, 64, 0, stream>>>(X, Y, angles, b, b > 0 ? 1 : 0);
    pass_high<<<ROWLEN / 2, 64, 0, stream>>>(Y, X, angles, b);
  }
  finalize_kernel<<<NSTATE / 256, 256, 0, stream>>>(X, (float*)d_out);
}